// EdgeConv_17609365914509
// MI455X (gfx1250) — compile-verified
//
#include <hip/hip_runtime.h>

typedef __attribute__((ext_vector_type(2))) float v2f;
typedef __attribute__((ext_vector_type(8))) float v8f;

#define CIN  64
#define XSTRIDE 132           // 128 + 4 pad: banks (4m + c) % 64 distinct for m=0..15
#define GEMM_WAVES 4          // 4 waves * 16*132 floats = 33.8 KB LDS

// ---------------------------------------------------------------------------
// Kernel 1: zero the accumulator (V*64 f32) and degree counts (V i32)
// ---------------------------------------------------------------------------
__global__ void zero_ws_kernel(float* __restrict__ acc, int* __restrict__ cnt,
                               int nacc, int nv) {
    int i = blockIdx.x * blockDim.x + threadIdx.x;
    if (i < nacc) acc[i] = 0.0f;
    if (i < nv)   cnt[i] = 0;
}

// ---------------------------------------------------------------------------
// Kernel 2: segment-sum of gathered neighbor features.
// One wave per chunk of EPW edges; lane owns channels [2*lane, 2*lane+1].
// tgt is sorted -> accumulate runs in registers, flush with f32 atomics.
// Indices are loaded 32-at-a-time and broadcast with __shfl (wave32).
// ---------------------------------------------------------------------------
__global__ void edge_agg_kernel(const float* __restrict__ feat,
                                const int*   __restrict__ src,
                                const int*   __restrict__ tgt,
                                float*       __restrict__ acc,
                                int*         __restrict__ cnt,
                                int E, int epw) {
    const int wave = blockIdx.x * (blockDim.x >> 5) + (threadIdx.x >> 5);
    const int lane = threadIdx.x & 31;
    const int e0 = wave * epw;
    if (e0 >= E) return;
    int e1 = e0 + epw; if (e1 > E) e1 = E;

    float ax = 0.0f, ay = 0.0f;
    int runT = -1, runLen = 0;

    for (int base = e0; base < e1; base += 32) {
        const int nvalid = min(32, e1 - base);
        const int ei = base + lane;
        const int tl = (ei < e1) ? tgt[ei] : -1;
        const int sl = (ei < e1) ? src[ei] : 0;
        for (int i = 0; i < nvalid; ++i) {
            const int t = __shfl(tl, i);
            const int s = __shfl(sl, i);
            if (t != runT) {
                if (runLen > 0) {
                    float* p = acc + (size_t)runT * CIN + lane * 2;
                    __hip_atomic_fetch_add(p,     ax, __ATOMIC_RELAXED, __HIP_MEMORY_SCOPE_AGENT);
                    __hip_atomic_fetch_add(p + 1, ay, __ATOMIC_RELAXED, __HIP_MEMORY_SCOPE_AGENT);
                    if (lane == 0)
                        __hip_atomic_fetch_add(cnt + runT, runLen, __ATOMIC_RELAXED, __HIP_MEMORY_SCOPE_AGENT);
                }
                ax = 0.0f; ay = 0.0f; runLen = 0; runT = t;
            }
            const float2 f = *(const float2*)(feat + (size_t)s * CIN + lane * 2);
            ax += f.x; ay += f.y; ++runLen;
        }
    }
    if (runLen > 0) {
        float* p = acc + (size_t)runT * CIN + lane * 2;
        __hip_atomic_fetch_add(p,     ax, __ATOMIC_RELAXED, __HIP_MEMORY_SCOPE_AGENT);
        __hip_atomic_fetch_add(p + 1, ay, __ATOMIC_RELAXED, __HIP_MEMORY_SCOPE_AGENT);
        if (lane == 0)
            __hip_atomic_fetch_add(cnt + runT, runLen, __ATOMIC_RELAXED, __HIP_MEMORY_SCOPE_AGENT);
    }
}

// ---------------------------------------------------------------------------
// Kernel 3: out[v,:] = W1*f_v + W2*(acc_v/cnt_v - f_v) + b  (0 if cnt==0)
// as a (16 x 128) @ (128 x 64) slab GEMM per wave using V_WMMA_F32_16X16X4_F32.
// X slab is staged in LDS (row stride 132 floats to avoid bank conflicts);
// B fragments stream from the 32KB weight matrix (L2/WGP$-resident).
// ---------------------------------------------------------------------------
__global__ void __launch_bounds__(GEMM_WAVES * 32)
fused_gemm_kernel(const float* __restrict__ feat,
                  const float* __restrict__ acc,
                  const int*   __restrict__ cnt,
                  const float* __restrict__ W,   // (64, 128) row-major
                  const float* __restrict__ b,   // (64,)
                  float*       __restrict__ out, // (V, 64)
                  int nslab, int Vtot) {
    __shared__ float X[GEMM_WAVES][16 * XSTRIDE];

    const int waveId = threadIdx.x >> 5;
    const int lane   = threadIdx.x & 31;
    const int slab   = blockIdx.x * GEMM_WAVES + waveId;
    if (slab >= nslab) return;            // whole-wave exit: EXEC stays all-1s

    const int v0 = slab * 16;
    const int m  = lane & 15;             // row (A) / col (B,D) within tile
    const int h  = lane >> 4;             // half: selects K pair / M half
    float* Xw = &X[waveId][0];

    // ---- stage X[16][128] into LDS: [f_v | acc_v/cnt_v - f_v] ----
    for (int r = 0; r < 16; ++r) {
        int v = v0 + r; if (v >= Vtot) v = Vtot - 1;
        const float2 f = *(const float2*)(feat + (size_t)v * CIN + lane * 2);
        const float2 a = *(const float2*)(acc  + (size_t)v * CIN + lane * 2);
        const int    c = cnt[v];
        const float inv = (c > 0) ? (1.0f / (float)c) : 0.0f;
        Xw[r * XSTRIDE + lane * 2 + 0]      = f.x;
        Xw[r * XSTRIDE + lane * 2 + 1]      = f.y;
        Xw[r * XSTRIDE + 64 + lane * 2 + 0] = a.x * inv - f.x;
        Xw[r * XSTRIDE + 64 + lane * 2 + 1] = a.y * inv - f.y;
    }
    // producer == consumer wave: DScnt dependency handled by compiler waits.

    v8f C0 = {}, C1 = {}, C2 = {}, C3 = {};
    for (int k0 = 0; k0 < 128; k0 += 4) {
        const int kk = k0 + 2 * h;
        // A fragment: lane -> (M=m, K=kk..kk+1)
        const v2f A = *(const v2f*)(Xw + m * XSTRIDE + kk);
        // B fragments: lane -> (K=kk..kk+1, N=nt*16+m); Wt[k][n] = W[n*128+k]
        const float* Wp = W + kk;
        const v2f B0 = *(const v2f*)(Wp + (size_t)(0  + m) * 128);
        const v2f B1 = *(const v2f*)(Wp + (size_t)(16 + m) * 128);
        const v2f B2 = *(const v2f*)(Wp + (size_t)(32 + m) * 128);
        const v2f B3 = *(const v2f*)(Wp + (size_t)(48 + m) * 128);
        C0 = __builtin_amdgcn_wmma_f32_16x16x4_f32(false, A, false, B0, (short)0, C0, false, false);
        C1 = __builtin_amdgcn_wmma_f32_16x16x4_f32(false, A, false, B1, (short)0, C1, false, false);
        C2 = __builtin_amdgcn_wmma_f32_16x16x4_f32(false, A, false, B2, (short)0, C2, false, false);
        C3 = __builtin_amdgcn_wmma_f32_16x16x4_f32(false, A, false, B3, (short)0, C3, false, false);
    }

    // ---- store: C/D layout: VGPR j -> M = j + 8*h, N = m (per 16-col tile) ----
    const float bb0 = b[ 0 + m];
    const float bb1 = b[16 + m];
    const float bb2 = b[32 + m];
    const float bb3 = b[48 + m];
    for (int j = 0; j < 8; ++j) {
        const int v = v0 + 8 * h + j;
        if (v >= Vtot) continue;
        const bool nz = cnt[v] > 0;
        float* o = out + (size_t)v * 64;
        o[ 0 + m] = nz ? (C0[j] + bb0) : 0.0f;
        o[16 + m] = nz ? (C1[j] + bb1) : 0.0f;
        o[32 + m] = nz ? (C2[j] + bb2) : 0.0f;
        o[48 + m] = nz ? (C3[j] + bb3) : 0.0f;
    }
}

// ---------------------------------------------------------------------------
extern "C" void kernel_launch(void* const* d_in, const int* in_sizes, int n_in,
                              void* d_out, int out_size, void* d_ws, size_t ws_size,
                              hipStream_t stream) {
    const float* feat = (const float*)d_in[0];
    const int*   src  = (const int*)  d_in[1];
    const int*   tgt  = (const int*)  d_in[2];
    const float* W    = (const float*)d_in[3];
    const float* b    = (const float*)d_in[4];
    float*       out  = (float*)d_out;

    const int V = in_sizes[0] / CIN;
    const int E = in_sizes[1];

    float* acc = (float*)d_ws;                                    // V*64 f32
    int*   cnt = (int*)((char*)d_ws + (size_t)V * CIN * sizeof(float)); // V i32

    // 1) zero accumulators
    {
        const int nacc = V * CIN;
        const int blk = 256;
        zero_ws_kernel<<<(nacc + blk - 1) / blk, blk, 0, stream>>>(acc, cnt, nacc, V);
    }
    // 2) edge aggregation (segment sums via run-accumulated f32 atomics)
    {
        const int EPW = 128;                       // edges per wave
        const int nwaves  = (E + EPW - 1) / EPW;
        const int nblocks = (nwaves + 7) / 8;      // 8 waves / 256-thread block
        edge_agg_kernel<<<nblocks, 256, 0, stream>>>(feat, src, tgt, acc, cnt, E, EPW);
    }
    // 3) fused normalize + linear via f32 WMMA
    {
        const int nslab = (V + 15) / 16;
        const int nblocks = (nslab + GEMM_WAVES - 1) / GEMM_WAVES;
        fused_gemm_kernel<<<nblocks, GEMM_WAVES * 32, 0, stream>>>(
            feat, acc, cnt, W, b, out, nslab, V);
    }
}